// IsometryReg_84920093377378
// MI455X (gfx1250) — compile-verified
//
#include <hip/hip_runtime.h>
#include <math.h>

typedef float v2f __attribute__((ext_vector_type(2)));
typedef float v8f __attribute__((ext_vector_type(8)));

#define NCLS  10
#define NPAD  16
#define NDIM  3072
#define BATCH 1024
#define TILE_M 16
#define KWAVES 8                 // waves per block (K-split)
#define KSLICE (NDIM / KWAVES)   // 384 K-elements per wave
#define EPS_REG 0.1f
#define NSTAB 1e-4f

// ws layout (floats):
//  [0, 49152)      Wpad [3072][16]
//  [49152, 49408)  M = W^T W  [16][16] (rows/cols 10..15 zero)
//  [49408, 50432)  per-sample reg [1024]
#define WS_WPAD 0
#define WS_M    49152
#define WS_REG  49408

__global__ void pad_w_kernel(const float* __restrict__ W, float* __restrict__ wpad) {
    int idx = blockIdx.x * blockDim.x + threadIdx.x;
    if (idx >= NDIM * NPAD) return;
    int k = idx >> 4, j = idx & 15;
    wpad[idx] = (j < NCLS) ? W[k * NCLS + j] : 0.0f;
}

// One wave: M = W^T W via V_WMMA_F32_16X16X4_F32.
// A[m,k] = W[kb+k, m], B[k,n] = W[kb+k, n]; with m = n = lane&15 and
// K = vgpr + 2*(lane>=16) for both operands, the per-lane data is identical.
__global__ void gram_kernel(const float* __restrict__ wpad, float* __restrict__ Mbuf) {
    int lane = threadIdx.x;
    int half = lane >> 4;
    int nn   = lane & 15;
    v8f c = {};
    for (int kb = 0; kb < NDIM; kb += 4) {
        int k0 = kb + 2 * half;
        v2f a;
        a.x = wpad[k0 * NPAD + nn];
        a.y = wpad[(k0 + 1) * NPAD + nn];
        c = __builtin_amdgcn_wmma_f32_16x16x4_f32(false, a, false, a,
                                                  (short)0, c, false, false);
    }
    // C layout: VGPR v -> row v (lanes 0-15) / row v+8 (lanes 16-31), col = lane&15
    #pragma unroll
    for (int v = 0; v < 8; ++v)
        Mbuf[(v + 8 * half) * NPAD + nn] = c[v];
}

__global__ void __launch_bounds__(KWAVES * 32)
iso_main_kernel(const float* __restrict__ data,
                const float* __restrict__ wpad,
                const float* __restrict__ bvec,
                const float* __restrict__ Mbuf,
                float* __restrict__ regout) {
    __shared__ float ldsPart[KWAVES * TILE_M * NPAD];  // 8 partial C tiles
    __shared__ float ldsLogits[TILE_M * NPAD];
    __shared__ float ldsM[NPAD * NPAD];

    int tid  = threadIdx.x;
    int wave = tid >> 5;
    int lane = tid & 31;
    int half = lane >> 4;
    int nn   = lane & 15;
    int tile = blockIdx.x;

    if (tid < NPAD * NPAD) ldsM[tid] = Mbuf[tid];

    // ---- partial logits: each wave accumulates a disjoint 384-wide K slice ----
    // A lane role: M-row = lane&15, float2 load at K = kbase + kb + 2*half
    const float* arow = data + ((size_t)tile * TILE_M + nn) * NDIM
                             + (size_t)wave * KSLICE + 2 * half;
    const float* wrow = wpad + (size_t)wave * KSLICE * NPAD;
    v8f c = {};
    #pragma unroll 4
    for (int kb = 0; kb < KSLICE; kb += 4) {
        v2f a = *(const v2f*)(arow + kb);
        int k0 = kb + 2 * half;
        v2f b;
        b.x = wrow[k0 * NPAD + nn];
        b.y = wrow[(k0 + 1) * NPAD + nn];
        c = __builtin_amdgcn_wmma_f32_16x16x4_f32(false, a, false, b,
                                                  (short)0, c, false, false);
    }
    #pragma unroll
    for (int v = 0; v < 8; ++v)
        ldsPart[wave * (TILE_M * NPAD) + (v + 8 * half) * NPAD + nn] = c[v];
    __syncthreads();

    // ---- reduce the 8 partial tiles (fixed order -> deterministic) ----
    {
        float acc = 0.0f;
        #pragma unroll
        for (int w = 0; w < KWAVES; ++w) acc += ldsPart[w * (TILE_M * NPAD) + tid];
        ldsLogits[tid] = acc;   // tid < 256 == TILE_M*NPAD
    }
    __syncthreads();

    // ---- per-sample closed-form isometry regularizer (threads 0..15) ----
    if (tid < TILE_M) {
        const float s = 1.0f - (float)NCLS * NSTAB;
        float lg[NCLS];
        #pragma unroll
        for (int k = 0; k < NCLS; ++k) lg[k] = ldsLogits[tid * NPAD + k] + bvec[k];

        float mx = lg[0];
        #pragma unroll
        for (int k = 1; k < NCLS; ++k) mx = fmaxf(mx, lg[k]);
        float q[NCLS], sum = 0.0f;
        #pragma unroll
        for (int k = 0; k < NCLS; ++k) { q[k] = expf(lg[k] - mx); sum += q[k]; }
        float inv = 1.0f / sum;

        float p, r[NCLS], g[NCLS], sumr = 0.0f;
        #pragma unroll
        for (int k = 0; k < NCLS; ++k) {
            q[k] *= inv;
            p = s * q[k] + NSTAB;
            r[k] = sqrtf(p);
            g[k] = s * q[k] / (2.0f * r[k]);     // dr_k/dl_k diagonal scale
            sumr += r[k];
        }
        float u = 1.0f / (1.0f - r[NCLS - 1]);

        // Row i of A (9x10): a_i*e_i + c_i*e_9 + d_i*q
        float av[NCLS - 1], cv[NCLS - 1], dv[NCLS - 1];
        #pragma unroll
        for (int i = 0; i < NCLS - 1; ++i) {
            av[i] = 2.0f * u * g[i];
            cv[i] = 2.0f * r[i] * u * u * g[NCLS - 1];
            dv[i] = -(av[i] + cv[i]);
        }

        // v = M q, alpha = q^T M q
        float vv[NCLS], alpha = 0.0f;
        #pragma unroll
        for (int k = 0; k < NCLS; ++k) {
            float acc = 0.0f;
            #pragma unroll
            for (int j = 0; j < NCLS; ++j) acc += ldsM[k * NPAD + j] * q[j];
            vv[k] = acc;
            alpha += q[k] * acc;
        }

        float arg    = fminf(sumr * 0.31622776601683794f, 1.0f);   // sum r / sqrt(10)
        float delta  = 2.0f * acosf(arg);
        float coeff  = 4.0f * (1.0f - r[NCLS - 1]) * (1.0f - r[NCLS - 1]);
        float factor = delta * delta / (coeff * EPS_REG * EPS_REG);

        float M99 = ldsM[9 * NPAD + 9], v9 = vv[9];
        float ss = 0.0f;
        #pragma unroll
        for (int i = 0; i < NCLS - 1; ++i) {
            #pragma unroll
            for (int j = 0; j < NCLS - 1; ++j) {
                float Gij = av[i] * av[j] * ldsM[i * NPAD + j]
                          + av[i] * cv[j] * ldsM[i * NPAD + 9]
                          + cv[i] * av[j] * ldsM[9 * NPAD + j]
                          + cv[i] * cv[j] * M99
                          + av[i] * dv[j] * vv[i]
                          + dv[i] * av[j] * vv[j]
                          + (cv[i] * dv[j] + dv[i] * cv[j]) * v9
                          + dv[i] * dv[j] * alpha;
                float d = Gij - ((i == j) ? factor : 0.0f);
                ss += d * d;
            }
        }
        regout[tile * TILE_M + tid] = sqrtf(ss) / (float)NDIM;
    }
}

__global__ void reduce_kernel(const float* __restrict__ regs, float* __restrict__ out) {
    __shared__ float buf[256];
    int t = threadIdx.x;
    buf[t] = regs[t] + regs[t + 256] + regs[t + 512] + regs[t + 768];
    __syncthreads();
    for (int w = 128; w > 0; w >>= 1) {
        if (t < w) buf[t] += buf[t + w];
        __syncthreads();
    }
    if (t == 0) {
        out[0] = buf[0] * (1.0f / (float)BATCH);
        out[1] = 0.0f;
    }
}

extern "C" void kernel_launch(void* const* d_in, const int* in_sizes, int n_in,
                              void* d_out, int out_size, void* d_ws, size_t ws_size,
                              hipStream_t stream) {
    const float* data = (const float*)d_in[0];   // [1024,3,32,32] f32
    const float* W    = (const float*)d_in[1];   // [3072,10] f32
    const float* b    = (const float*)d_in[2];   // [10] f32

    float* ws   = (float*)d_ws;
    float* wpad = ws + WS_WPAD;
    float* Mbuf = ws + WS_M;
    float* regs = ws + WS_REG;

    pad_w_kernel<<<(NDIM * NPAD + 255) / 256, 256, 0, stream>>>(W, wpad);
    gram_kernel<<<1, 32, 0, stream>>>(wpad, Mbuf);
    iso_main_kernel<<<BATCH / TILE_M, KWAVES * 32, 0, stream>>>(data, wpad, b, Mbuf, regs);
    reduce_kernel<<<1, 256, 0, stream>>>(regs, (float*)d_out);
}